// bow_labeler_40870908789454
// MI455X (gfx1250) — compile-verified
//
#include <hip/hip_runtime.h>
#include <hip/hip_bf16.h>

// Shapes from the reference
#define Bsz 256
#define Lsz 512
#define Dsz 768
#define NOUT 54      // 13*4 + 2
#define NPAD 64      // padded to 4 N-tiles of 16

typedef __attribute__((ext_vector_type(2))) float v2f;
typedef __attribute__((ext_vector_type(8))) float v8f;

// ---------------------------------------------------------------------------
// Kernel 1: pack W13/W14/b13/b14 into a zero-padded [64 x 768] weight matrix
// and a [64] bias vector so the WMMA GEMM needs no conditionals in the K loop.
// Row n (n<52) = W13 flat row n (W13 is [13,4,768] -> rows contiguous),
// rows 52..53 = W14, rows 54..63 = 0.
// ---------------------------------------------------------------------------
__global__ __launch_bounds__(256) void pack_kernel(
    const float* __restrict__ W13, const float* __restrict__ b13,
    const float* __restrict__ W14, const float* __restrict__ b14,
    float* __restrict__ Wp, float* __restrict__ biasp) {
  int idx = blockIdx.x * 256 + threadIdx.x;
  if (idx < NPAD * Dsz) {
    int n = idx / Dsz;
    int d = idx - n * Dsz;
    float v = 0.0f;
    if (n < 52)      v = W13[n * Dsz + d];
    else if (n < 54) v = W14[(n - 52) * Dsz + d];
    Wp[idx] = v;
  }
  if (idx < NPAD) {
    float bv = 0.0f;
    if (idx < 52)      bv = b13[idx];
    else if (idx < 54) bv = b14[idx - 52];
    biasp[idx] = bv;
  }
}

// ---------------------------------------------------------------------------
// Kernel 2: masked mean pooling (the bandwidth-bound stage, ~403 MB streamed).
// grid = B*3 blocks of 256 threads; block -> (batch b, 256-wide D chunk).
// Mask row cached in LDS; hidden streamed with non-temporal loads (touched
// exactly once, larger than L2 -> don't pollute).
// ---------------------------------------------------------------------------
__global__ __launch_bounds__(256) void pool_kernel(
    const float* __restrict__ hidden, const int* __restrict__ mask,
    float* __restrict__ pooled) {
  int b     = blockIdx.x / 3;
  int chunk = blockIdx.x % 3;
  int tid   = threadIdx.x;

  __shared__ float smask[Lsz];
  const int* mrow = mask + b * Lsz;
  for (int i = tid; i < Lsz; i += 256) smask[i] = (float)mrow[i];
  __syncthreads();

  int d = chunk * 256 + tid;
  const float* hb = hidden + (size_t)b * Lsz * Dsz + d;

  float acc = 0.0f;
  float cnt = 0.0f;
#pragma unroll 4
  for (int l = 0; l < Lsz; ++l) {
    float m = smask[l];
    cnt += m;
    acc += m * __builtin_nontemporal_load(hb + (size_t)l * Dsz);
  }
  pooled[b * Dsz + d] = acc / cnt;  // reference divides too (NaN if cnt==0 in both)
}

// ---------------------------------------------------------------------------
// Kernel 3: out = pooled[256x768] @ Wp^T[768x64] + bias, via
// V_WMMA_F32_16X16X4_F32. One wave per 16x16 output tile; K=768 -> 192 WMMAs.
//
// fp32 16x16x4 layouts (ISA 7.12.2):
//   A 16x4 : lane&15 = M, VGPR j (j=0,1): K = 2*(lane>>4) + j      -> float2 load
//   B 4x16 : lane&15 = N, VGPR j:         K = 2*(lane>>4) + j      -> float2 load
//   C/D    : VGPR r: M = r + 8*(lane>>4), N = lane&15
// EXEC is all-1s through the WMMA loop (no divergence until epilogue).
// ---------------------------------------------------------------------------
__global__ __launch_bounds__(32) void gemm_wmma(
    const float* __restrict__ pooled, const float* __restrict__ Wp,
    const float* __restrict__ biasp, float* __restrict__ out) {
  int mt   = blockIdx.x;          // 0..15  (M tiles of 256)
  int nt   = blockIdx.y;          // 0..3   (N tiles of 64)
  int lane = threadIdx.x;         // 0..31

  int row16 = lane & 15;          // M for A, N for B
  int khalf = (lane >> 4) * 2;    // K sub-offset per lane half

  const float* Arow = pooled + (size_t)(mt * 16 + row16) * Dsz + khalf;
  const float* Brow = Wp     + (size_t)(nt * 16 + row16) * Dsz + khalf;

  v8f acc = {};
  for (int k0 = 0; k0 < Dsz; k0 += 4) {
    v2f a = *(const v2f*)(Arow + k0);
    v2f b = *(const v2f*)(Brow + k0);
    // (neg_a, A, neg_b, B, c_mod, C, reuse_a, reuse_b)
    acc = __builtin_amdgcn_wmma_f32_16x16x4_f32(
        false, a, false, b, (short)0, acc, false, false);
  }

  int n = nt * 16 + (lane & 15);
  if (n < NOUT) {
    float bias  = biasp[n];
    int   mbase = mt * 16 + ((lane >> 4) << 3);
#pragma unroll
    for (int r = 0; r < 8; ++r) {
      out[(size_t)(mbase + r) * NOUT + n] = acc[r] + bias;
    }
  }
}

// ---------------------------------------------------------------------------
extern "C" void kernel_launch(void* const* d_in, const int* in_sizes, int n_in,
                              void* d_out, int out_size, void* d_ws, size_t ws_size,
                              hipStream_t stream) {
  const float* hidden = (const float*)d_in[0];   // [256,512,768] f32
  const int*   mask   = (const int*)d_in[1];     // [256,512] i32
  const float* W13    = (const float*)d_in[2];   // [13,4,768] f32
  const float* b13    = (const float*)d_in[3];   // [13,4] f32
  const float* W14    = (const float*)d_in[4];   // [2,768] f32
  const float* b14    = (const float*)d_in[5];   // [2] f32
  float*       out    = (float*)d_out;           // [256,54] f32

  // Workspace layout
  float* pooled = (float*)d_ws;                          // 256*768 f32
  float* Wp     = pooled + (size_t)Bsz * Dsz;            // 64*768 f32
  float* biasp  = Wp + (size_t)NPAD * Dsz;               // 64 f32

  // Pack weights (independent of pooling; stream order gates the GEMM).
  pack_kernel<<<(NPAD * Dsz + 255) / 256, 256, 0, stream>>>(
      W13, b13, W14, b14, Wp, biasp);

  // Masked mean pooling: 768 blocks x 256 threads.
  pool_kernel<<<Bsz * 3, 256, 0, stream>>>(hidden, mask, pooled);

  // WMMA GEMM: 16 x 4 tiles, one wave each.
  gemm_wmma<<<dim3(16, 4), 32, 0, stream>>>(pooled, Wp, biasp, out);
}